// SSTModel_65970697666980
// MI455X (gfx1250) — compile-verified
//
#include <hip/hip_runtime.h>
#include <math.h>

#define N_FFT   2048
#define HOP     512
#define NFREQ   1025
#define NT      8193
#define SIG_LEN 4194304
#define PI_F    3.14159265358979323846f

typedef __attribute__((ext_vector_type(2))) float v2f;
typedef __attribute__((ext_vector_type(4))) float v4f;
typedef __attribute__((ext_vector_type(8))) float v8f;

// D(16x16,f32) = A(16x4,f32) * B(4x16,f32) + C ; full-precision f32 WMMA
static __device__ __forceinline__ v8f wmma4(v2f a, v2f b, v8f c) {
  return __builtin_amdgcn_wmma_f32_16x16x4_f32(false, a, false, b, (short)0, c,
                                               false, false);
}

// Hardware GLOBAL_ATOMIC_ADD_F32 (no CAS loop); scatter-add order is unordered
// in the reference too, so fp-reassociation across lanes is acceptable.
static __device__ __forceinline__ void atomic_fadd(float* p, float v) {
  unsafeAtomicAdd(p, v);
}

static __device__ __forceinline__ int reflect_idx(int m) {
  if (m < 0) m = -m;
  if (m >= SIG_LEN) m = 2 * SIG_LEN - 2 - m;
  return m;
}

// Wide zero-fill: GLOBAL_STORE_B128 for the bulk, scalar tail.
__global__ void zero_kernel(float* __restrict__ out, int n) {
  const int n4 = n >> 2;
  int i = blockIdx.x * blockDim.x + threadIdx.x;
  if (i < n4) {
    v4f z = {};
    *(v4f*)(out + 4 * (size_t)i) = z;
  }
  if (i < (n & 3)) out[(size_t)(n4 << 2) + i] = 0.0f;  // tail (threads 0..2 of block 0 also pass? no: i global)
}

// One workgroup per output time column t (t = 0 .. NT-2). Column NT-1 stays 0.
__global__ __launch_bounds__(256) void sst_col_kernel(const float* __restrict__ x,
                                                      float* __restrict__ out) {
  const int t    = blockIdx.x;
  const int tid  = threadIdx.x;
  const int lane = tid & 31;
  const int wave = tid >> 5;          // 8 waves

  __shared__ float Ar[N_FFT], Ai[N_FFT], Br[N_FFT], Bi[N_FFT];

  // ---- build packed complex input: z = win*frame_t + i*win*frame_{t+1} ----
  const int base = t * HOP - (N_FFT / 2);   // reflect-padded index -> x coords
  #pragma unroll
  for (int j = tid; j < N_FFT; j += 256) {
    const float w = 0.5f * (1.0f - __cosf((2.0f * PI_F / (float)N_FFT) * (float)j));
    Ar[j] = w * x[reflect_idx(base + j)];
    Ai[j] = w * x[reflect_idx(base + HOP + j)];
  }
  __syncthreads();

  // ---- DFT-16 matrix as WMMA A-operands (16x4 f32 layout), K-chunks j=0..3 ----
  // lanes 0-15: M=lane, VGPR0->K=4j, VGPR1->K=4j+1 ; lanes 16-31: M=lane-16, K=4j+2/4j+3
  const int Mrow = lane & 15;
  const int kHi  = (lane >> 4) << 1;  // 0 or 2
  v2f Fr[4], Fi[4], Fni[4];           // Re(F16), Im(F16), -Im(F16)
  #pragma unroll
  for (int j = 0; j < 4; ++j) {
    const int k0 = 4 * j + kHi;
    float s0, c0, s1, c1;
    __sincosf((2.0f * PI_F / 16.0f) * (float)((Mrow * k0) & 15), &s0, &c0);
    __sincosf((2.0f * PI_F / 16.0f) * (float)((Mrow * (k0 + 1)) & 15), &s1, &c1);
    Fr[j].x  = c0;  Fr[j].y  = c1;
    Fi[j].x  = -s0; Fi[j].y  = -s1;   // W16 = exp(-i*2pi*M*K/16)
    Fni[j].x = s0;  Fni[j].y = s1;
  }

  const int colLocal = lane & 15;
  const int rowHalf  = (lane >> 4) << 3;   // D regs cover M=r (lanes<16) / M=r+8

  // ---- Stage 1: G[k1][n2] = sum_n1 F16[k1][n1]*z[n1*128+n2], tw W_2048^{k1 n2} ----
  {
    const int n2 = wave * 16 + colLocal;
    v8f Cr = {}, Ci = {};
    #pragma unroll
    for (int j = 0; j < 4; ++j) {
      const int kk = 4 * j + kHi;     // B rows (K) per documented layout
      v2f br, bi;
      br.x = Ar[kk * 128 + n2];       br.y = Ar[(kk + 1) * 128 + n2];
      bi.x = Ai[kk * 128 + n2];       bi.y = Ai[(kk + 1) * 128 + n2];
      Cr = wmma4(Fr[j],  br, Cr);     // Re += Fr*Br
      Cr = wmma4(Fni[j], bi, Cr);     // Re += (-Fi)*Bi
      Ci = wmma4(Fr[j],  bi, Ci);     // Im += Fr*Bi
      Ci = wmma4(Fi[j],  br, Ci);     // Im += Fi*Br
    }
    #pragma unroll
    for (int r = 0; r < 8; ++r) {
      const int k1 = r + rowHalf;
      float s, c;
      __sincosf((2.0f * PI_F / 2048.0f) * (float)(k1 * n2), &s, &c);
      const float a = Cr[r], b = Ci[r];      // * (c - i s)
      Br[k1 * 128 + n2] = a * c + b * s;
      Bi[k1 * 128 + n2] = b * c - a * s;
    }
  }
  __syncthreads();

  // ---- Stage 2: per k1 row, H[q1][m2] = sum_m1 F16[q1][m1]*G'[k1][8*m1+m2] ----
  {
    const int c2 = wave * 16 + colLocal;   // batched columns = (k1, m2)
    const int k1 = c2 >> 3, m2 = c2 & 7;
    v8f Cr = {}, Ci = {};
    #pragma unroll
    for (int j = 0; j < 4; ++j) {
      const int m1 = 4 * j + kHi;
      v2f br, bi;
      br.x = Br[k1 * 128 + 8 * m1 + m2];   br.y = Br[k1 * 128 + 8 * (m1 + 1) + m2];
      bi.x = Bi[k1 * 128 + 8 * m1 + m2];   bi.y = Bi[k1 * 128 + 8 * (m1 + 1) + m2];
      Cr = wmma4(Fr[j],  br, Cr);
      Cr = wmma4(Fni[j], bi, Cr);
      Ci = wmma4(Fr[j],  bi, Ci);
      Ci = wmma4(Fi[j],  br, Ci);
    }
    #pragma unroll
    for (int r = 0; r < 8; ++r) {
      const int q1 = r + rowHalf;
      float s, c;
      __sincosf((2.0f * PI_F / 128.0f) * (float)(q1 * m2), &s, &c);
      const float a = Cr[r], b = Ci[r];
      Ar[k1 * 128 + q1 * 8 + m2] = a * c + b * s;
      Ai[k1 * 128 + q1 * 8 + m2] = b * c - a * s;
    }
  }
  __syncthreads();

  // ---- Stage 3: 256 radix-8 DFTs (VALU), output X[k1 + 16*q1 + 256*q2] ----
  {
    const int k1 = tid >> 4, q1 = tid & 15;
    float inr[8], ini[8];
    #pragma unroll
    for (int m = 0; m < 8; ++m) {
      inr[m] = Ar[k1 * 128 + q1 * 8 + m];
      ini[m] = Ai[k1 * 128 + q1 * 8 + m];
    }
    const float R2 = 0.70710678118654752440f;
    const float C8[8] = {1.0f, R2, 0.0f, -R2, -1.0f, -R2, 0.0f, R2};
    const float S8[8] = {0.0f, R2, 1.0f, R2, 0.0f, -R2, -1.0f, -R2};
    #pragma unroll
    for (int q2 = 0; q2 < 8; ++q2) {
      float sr = 0.0f, si = 0.0f;
      #pragma unroll
      for (int m = 0; m < 8; ++m) {
        const int idx = (m * q2) & 7;          // W8^{m*q2} = C - iS
        sr += inr[m] * C8[idx] + ini[m] * S8[idx];
        si += ini[m] * C8[idx] - inr[m] * S8[idx];
      }
      const int kout = k1 + 16 * q1 + 256 * q2;
      Br[kout] = sr;
      Bi[kout] = si;
    }
  }
  __syncthreads();

  // ---- unpack the two real spectra, phase diff, trunc, scatter (L2 atomics) ----
  for (int f = tid; f < NFREQ; f += 256) {
    const int m = (N_FFT - f) & (N_FFT - 1);
    const float zr1 = Br[f], zi1 = Bi[f];
    const float zr2 = Br[m], zi2 = Bi[m];
    // S_t = (Z[f] + conj(Z[-f]))/2 ; S_{t+1} = -i*(Z[f] - conj(Z[-f]))/2
    const float str = 0.5f * (zr1 + zr2);
    const float sti = 0.5f * (zi1 - zi2);
    const float s1r = 0.5f * (zi1 + zi2);
    const float s1i = 0.5f * (zr2 - zr1);
    const float d   = atan2f(s1i, s1r) - atan2f(sti, str);
    const int   kb  = (int)truncf((float)f + d);
    if (kb >= 0 && kb < NFREQ) {
      float* p = out + ((size_t)kb * NT + (size_t)t) * 2;
      atomic_fadd(p,     str);
      atomic_fadd(p + 1, sti);
    }
  }
}

extern "C" void kernel_launch(void* const* d_in, const int* in_sizes, int n_in,
                              void* d_out, int out_size, void* d_ws, size_t ws_size,
                              hipStream_t stream) {
  (void)in_sizes; (void)n_in; (void)d_ws; (void)ws_size;
  const float* x = (const float*)d_in[0];
  float* out = (float*)d_out;

  // output (incl. untouched last column) must start at zero every call
  const int n4 = out_size >> 2;
  zero_kernel<<<(n4 + 255) / 256, 256, 0, stream>>>(out, out_size);
  // one workgroup per time column 0..NT-2
  sst_col_kernel<<<NT - 1, 256, 0, stream>>>(x, out);
}